// PIT_LFMMILoss_23398981829345
// MI455X (gfx1250) — compile-verified
//
#include <hip/hip_runtime.h>
#include <stdint.h>

// ---------------------------------------------------------------------------
// PIT-LFMMI loss, forward-only, for the fixed setup:
//   B=8, S=2, T=512, C=2048, den: Nd=1024/Ad=16384, num: Nn=256/An=2048, P=2
// One block per FSM-sequence. alpha + segment accumulators in LDS, arc data
// in registers, llh rows double-buffered via gfx1250 async-to-LDS (ASYNCcnt).
// 3 barriers per time step.
// ---------------------------------------------------------------------------

#define NEGV (-1e30f)

// Order-preserving float -> u32 map so segment-max can use native ds_max_u32.
__device__ __forceinline__ uint32_t fmap(float f) {
  uint32_t b = __float_as_uint(f);
  return (b & 0x80000000u) ? ~b : (b | 0x80000000u);
}
__device__ __forceinline__ float funmap(uint32_t u) {
  return __uint_as_float((u & 0x80000000u) ? (u & 0x7fffffffu) : ~u);
}
#define FMAP_NEGINF 0x007FFFFFu  // fmap(-inf)

// Generic (flat) pointer to a __shared__ object keeps the wave-relative LDS
// byte offset in its low 32 bits (CDNA5 aperture layout).
__device__ __forceinline__ uint32_t lds_off(const void* p) {
  return (uint32_t)(uintptr_t)p;
}

// gfx1250 async global->LDS copy, SADDR form: vdst = LDS byte offset,
// vaddr = 32-bit per-lane offset, saddr = 64-bit base. Tracked by ASYNCcnt.
__device__ __forceinline__ void async_ld_b128(uint32_t ldsaddr, uint64_t base,
                                              uint32_t voff) {
  asm volatile("global_load_async_to_lds_b128 %0, %1, %2"
               :: "v"(ldsaddr), "v"(voff), "s"(base)
               : "memory");
}
template <int N> __device__ __forceinline__ void wait_async() {
  if constexpr (N <= 0)      asm volatile("s_wait_asynccnt 0" ::: "memory");
  else if constexpr (N == 1) asm volatile("s_wait_asynccnt 1" ::: "memory");
  else                       asm volatile("s_wait_asynccnt 2" ::: "memory");
}

// ---------------------------------------------------------------------------
// One block = one FSM forward pass over T frames.
//   NST == NT (one state per thread), APT = NARC/NT arcs per thread.
// NUMMODE=false: shared denominator graph, block id = s*B+b.
// NUMMODE=true : per-(s,b) numerator graph, block id = p*(S*B)+s*B+b, llh
//                taken from permuted speaker q (S==2).
// ---------------------------------------------------------------------------
template <int NST, int NARC, int NT, int CDIM, bool NUMMODE, int WAITK>
__global__ __launch_bounds__(NT)
void fsm_fwd(const float* __restrict__ llhs,     // [B,S,T,C]
             const int*   __restrict__ seqlens,  // [B,S]
             const int*   __restrict__ asrc, const int* __restrict__ adst,
             const int*   __restrict__ apdf, const float* __restrict__ awt,
             const float* __restrict__ init0, const float* __restrict__ fin0,
             float* __restrict__ out, int B, int S, int T)
{
  static_assert(NST == NT, "one state per thread");
  constexpr int APT = NARC / NT;
  static_assert((APT & 3) == 0, "vectorized arc load needs APT % 4 == 0");

  __shared__ float    alpha[NST];
  __shared__ uint32_t mbuf[NST];
  __shared__ float    sbuf[NST];
  __shared__ float    llbuf[2][CDIM];

  const int tid = threadIdx.x;
  const int bid = blockIdx.x;

  int s, b, q, arcoff, stoff, outidx;
  if constexpr (!NUMMODE) {
    s = bid / B; b = bid % B; q = s;
    arcoff = 0; stoff = 0;
    outidx = s * B + b;
  } else {
    const int p = bid / (S * B);
    const int r = bid % (S * B);
    s = r / B; b = r % B;
    q = p ? (S - 1 - s) : s;              // speaker permutation (S==2)
    arcoff = (s * B + b) * NARC;
    stoff  = (s * B + b) * NST;
    outidx = p * S * B + s * B + b;
  }

  const int    seqlen = seqlens[b * S + q];
  const float* llh    = llhs + (size_t)(b * S + q) * (size_t)T * CDIM;

  // Arc data -> registers, vectorized b128 loads (once, reused all T steps).
  uint32_t ap[APT];
  float    aw[APT];
#pragma unroll
  for (int c = 0; c < APT / 4; ++c) {
    const int a = arcoff + (tid + c * NT) * 4;
    const int4   s4 = *(const int4*)  &asrc[a];
    const int4   d4 = *(const int4*)  &adst[a];
    const int4   p4 = *(const int4*)  &apdf[a];
    const float4 w4 = *(const float4*)&awt[a];
    ap[c*4+0] = (uint32_t)s4.x | ((uint32_t)d4.x << 10) | ((uint32_t)p4.x << 20);
    ap[c*4+1] = (uint32_t)s4.y | ((uint32_t)d4.y << 10) | ((uint32_t)p4.y << 20);
    ap[c*4+2] = (uint32_t)s4.z | ((uint32_t)d4.z << 10) | ((uint32_t)p4.z << 20);
    ap[c*4+3] = (uint32_t)s4.w | ((uint32_t)d4.w << 10) | ((uint32_t)p4.w << 20);
    aw[c*4+0] = w4.x; aw[c*4+1] = w4.y; aw[c*4+2] = w4.z; aw[c*4+3] = w4.w;
  }

  alpha[tid] = init0[stoff + tid];
  mbuf[tid]  = FMAP_NEGINF;               // pre-init accumulators (reset is
  sbuf[tid]  = 0.0f;                      // folded into the alpha-update pass)

  const uint32_t lb0 = lds_off(&llbuf[0][0]);
  const uint32_t lb1 = lds_off(&llbuf[1][0]);
  constexpr int ROWB = CDIM * 4;

  // Prime the pipeline: row 0 -> buffer 0.
  {
    const uint64_t g = (uint64_t)(uintptr_t)llh;
#pragma unroll
    for (int off = 0; off < ROWB; off += NT * 16)
      if (off + tid * 16 < ROWB)
        async_ld_b128(lb0 + off + tid * 16, g, (uint32_t)(off + tid * 16));
  }

  for (int t = 0; t < T; ++t) {
    const int cur = t & 1;
    if (t + 1 < T) {                      // prefetch next row, then wait cur.
      // issue-then-wait keeps <=2 rows in flight, so same-parity buffers
      // never have overlapping async writes (async done returns in order).
      const uint64_t g = (uint64_t)(uintptr_t)(llh + (size_t)(t + 1) * CDIM);
      const uint32_t lb = cur ? lb0 : lb1;
#pragma unroll
      for (int off = 0; off < ROWB; off += NT * 16)
        if (off + tid * 16 < ROWB)
          async_ld_b128(lb + off + tid * 16, g, (uint32_t)(off + tid * 16));
      wait_async<WAITK>();
    } else {
      wait_async<0>();
    }
    __syncthreads();  // [A] row t + previous alpha/accumulator writes visible

    if (t < seqlen) {
      const float* lrow = cur ? &llbuf[1][0] : &llbuf[0][0];

      float sc[APT];
#pragma unroll
      for (int k = 0; k < APT; ++k) {     // scores + segment max (ds_max_u32)
        const uint32_t pk = ap[k];
        const float v = alpha[pk & 1023u] + aw[k] + lrow[pk >> 20];
        sc[k] = v;
        atomicMax(&mbuf[(pk >> 10) & 1023u], fmap(v));
      }
      __syncthreads();  // [B]

#pragma unroll
      for (int k = 0; k < APT; ++k) {     // segment sum of exp (ds_add_f32)
        const uint32_t d = (ap[k] >> 10) & 1023u;
        const float mm = funmap(mbuf[d]);
        atomicAdd(&sbuf[d], __expf(sc[k] - mm));
      }
      __syncthreads();  // [C]

      {                                   // alpha update + accumulator reset
        const float mm = fmaxf(funmap(mbuf[tid]), NEGV);
        alpha[tid] = mm + __logf(fmaxf(sbuf[tid], 1e-30f));
        mbuf[tid]  = FMAP_NEGINF;
        sbuf[tid]  = 0.0f;
      }
      // no trailing barrier: [A] of the next iteration orders these writes
      // before the next max pass, and [B]/[C] already separate llh-buffer
      // reads from the next prefetch issue.
    }
  }

  // logsumexp(alpha + final); mbuf[0]/sbuf[0] are in reset state here.
  __syncthreads();
  const float v = alpha[tid] + fin0[stoff + tid];
  atomicMax(&mbuf[0], fmap(v));
  __syncthreads();
  const float mm = funmap(mbuf[0]);
  atomicAdd(&sbuf[0], __expf(v - mm));
  __syncthreads();
  if (tid == 0) out[outidx] = mm + __logf(sbuf[0]);
}

// loss = sum_b -( min_p sum_s num[p,s,b]  -  sum_s den[s,b] )
__global__ void pit_finalize(const float* __restrict__ den_part,
                             const float* __restrict__ num_part,
                             float* __restrict__ out, int B, int S, int P)
{
  if (threadIdx.x == 0 && blockIdx.x == 0) {
    float loss = 0.0f;
    for (int b = 0; b < B; ++b) {
      float den = 0.0f;
      for (int s = 0; s < S; ++s) den += den_part[s * B + b];
      float best = 3.4e38f;
      for (int p = 0; p < P; ++p) {
        float acc = 0.0f;
        for (int s = 0; s < S; ++s) acc += num_part[p * S * B + s * B + b];
        best = fminf(best, acc);
      }
      loss += -(best - den);
    }
    out[0] = loss;
  }
}

extern "C" void kernel_launch(void* const* d_in, const int* in_sizes, int n_in,
                              void* d_out, int out_size, void* d_ws,
                              size_t ws_size, hipStream_t stream)
{
  (void)in_sizes; (void)n_in; (void)out_size; (void)ws_size;

  constexpr int B = 8, S = 2, T = 512, C = 2048;
  constexpr int Nd = 1024, Ad = 16384;
  constexpr int Nn = 256,  An = 2048;
  constexpr int P = 2;

  const float* est    = (const float*)d_in[0];
  const int*   seql   = (const int*)  d_in[1];
  const int*   dsrc   = (const int*)  d_in[2];
  const int*   ddst   = (const int*)  d_in[3];
  const int*   dpdf   = (const int*)  d_in[4];
  const float* dwt    = (const float*)d_in[5];
  const float* dinit  = (const float*)d_in[6];
  const float* dfinal = (const float*)d_in[7];
  const int*   nsrc   = (const int*)  d_in[8];
  const int*   ndst   = (const int*)  d_in[9];
  const int*   npdf   = (const int*)  d_in[10];
  const float* nwt    = (const float*)d_in[11];
  const float* ninit  = (const float*)d_in[12];
  const float* nfinal = (const float*)d_in[13];

  float* den_part = (float*)d_ws;            // [S*B]
  float* num_part = den_part + S * B;        // [P*S*B]

  // Denominator: 16 blocks x 1024 threads; waves 0-15 issue 1 async row chunk each, waves 16-31 none.
  fsm_fwd<Nd, Ad, 1024, C, false, 1><<<S * B, 1024, 0, stream>>>(
      est, seql, dsrc, ddst, dpdf, dwt, dinit, dfinal, den_part, B, S, T);

  // Numerator: 32 blocks x 256 threads; each wave issues 2 async row chunks.
  fsm_fwd<Nn, An, 256, C, true, 2><<<P * S * B, 256, 0, stream>>>(
      est, seql, nsrc, ndst, npdf, nwt, ninit, nfinal, num_part, B, S, T);

  pit_finalize<<<1, 32, 0, stream>>>(den_part, num_part, (float*)d_out, B, S, P);
}